// HashEmbedder_1769526526582
// MI455X (gfx1250) — compile-verified
//
#include <hip/hip_runtime.h>
#include <stdint.h>

// Instant-NGP hash-grid encoder forward, tuned for MI455X (gfx1250):
//  - wave32, 256-thread blocks; fully unrolled level loop -> up to 128
//    outstanding global_load_b64 gathers per thread for latency hiding
//  - float2 ext-vector math -> packed v_pk_* f32 VALU
//  - tables (64 MB) kept L2-resident (192 MB L2): streaming output stores and
//    coordinate loads are marked non-temporal (TH=NT) so they don't evict
//    table cachelines; table gathers use default RT policy.

typedef __attribute__((ext_vector_type(2))) float f32x2;
typedef __attribute__((ext_vector_type(4))) float f32x4;

#define NPTS   1048576u
#define NLEV   16
#define LOG2T  19
#define TBL    (1u << LOG2T)
#define HMASK  (TBL - 1u)
#define P1     2654435761u
#define P2     805459861u

__global__ __launch_bounds__(256) void hashgrid_fwd(
    const float* __restrict__ x,       // [NPTS][3]
    const float* __restrict__ tables,  // [NLEV][TBL][2]
    float* __restrict__ out,           // [NPTS][NLEV*2]
    float* __restrict__ keep)          // [NPTS]
{
  const uint32_t p = blockIdx.x * 256u + threadIdx.x;
  if (p >= NPTS) return;

  // Streaming, read-once coordinate loads: non-temporal.
  const float px = __builtin_nontemporal_load(x + 3u * p + 0u);
  const float py = __builtin_nontemporal_load(x + 3u * p + 1u);
  const float pz = __builtin_nontemporal_load(x + 3u * p + 2u);

  const bool inbox = (px >= 0.0f) & (px <= 1.0f) &
                     (py >= 0.0f) & (py <= 1.0f) &
                     (pz >= 0.0f) & (pz <= 1.0f);
  __builtin_nontemporal_store(inbox ? 1.0f : 0.0f, keep + p);

  const float cx = fminf(fmaxf(px, 0.0f), 1.0f);
  const float cy = fminf(fmaxf(py, 0.0f), 1.0f);
  const float cz = fminf(fmaxf(pz, 0.0f), 1.0f);

  // floor(16 * 2^(l/3)) for l = 0..15, matches reference fp32 computation.
  const float kRes[NLEV] = {16.f, 20.f, 25.f, 32.f, 40.f, 50.f, 64.f, 80.f,
                            101.f, 128.f, 161.f, 203.f, 256.f, 322.f, 406.f, 512.f};

  f32x2 resv[NLEV];

#pragma unroll
  for (int l = 0; l < NLEV; ++l) {
    const float r = kRes[l];
    const float fx = cx * r, fy = cy * r, fz = cz * r;
    const float bxf = floorf(fx), byf = floorf(fy), bzf = floorf(fz);
    const float wx = fx - bxf, wy = fy - byf, wz = fz - bzf;
    const float ox = 1.0f - wx, oy = 1.0f - wy, oz = 1.0f - wz;

    const uint32_t bx = (uint32_t)bxf;
    const uint32_t by = (uint32_t)byf;
    const uint32_t bz = (uint32_t)bzf;

    // Spatial hash: h = x ^ (y*P1) ^ (z*P2), factored (2 muls + 3 adds + 12 xors).
    const uint32_t hy0 = by * P1, hy1 = hy0 + P1;
    const uint32_t hz0 = bz * P2, hz1 = hz0 + P2;
    const uint32_t x0y0 = bx ^ hy0;
    const uint32_t x0y1 = bx ^ hy1;
    const uint32_t x1y0 = (bx + 1u) ^ hy0;
    const uint32_t x1y1 = (bx + 1u) ^ hy1;

    const float* tb = tables + (size_t)l * (size_t)(TBL * 2u);

    // 8 independent 8-byte gathers (global_load_b64); default RT policy so
    // table lines stay resident in L2.
    const f32x2 e000 = *(const f32x2*)(tb + 2u * ((x0y0 ^ hz0) & HMASK));
    const f32x2 e001 = *(const f32x2*)(tb + 2u * ((x0y0 ^ hz1) & HMASK));
    const f32x2 e010 = *(const f32x2*)(tb + 2u * ((x0y1 ^ hz0) & HMASK));
    const f32x2 e011 = *(const f32x2*)(tb + 2u * ((x0y1 ^ hz1) & HMASK));
    const f32x2 e100 = *(const f32x2*)(tb + 2u * ((x1y0 ^ hz0) & HMASK));
    const f32x2 e101 = *(const f32x2*)(tb + 2u * ((x1y0 ^ hz1) & HMASK));
    const f32x2 e110 = *(const f32x2*)(tb + 2u * ((x1y1 ^ hz0) & HMASK));
    const f32x2 e111 = *(const f32x2*)(tb + 2u * ((x1y1 ^ hz1) & HMASK));

    // Trilinear interpolation, 2 features packed per op (v_pk_* f32 path).
    const f32x2 c00 = e000 * ox + e100 * wx;
    const f32x2 c01 = e001 * ox + e101 * wx;
    const f32x2 c10 = e010 * ox + e110 * wx;
    const f32x2 c11 = e011 * ox + e111 * wx;
    const f32x2 c0  = c00 * oy + c10 * wy;
    const f32x2 c1  = c01 * oy + c11 * wy;
    resv[l] = c0 * oz + c1 * wz;
  }

  // 8 x global_store_b128 (non-temporal): one contiguous 128-byte block per
  // point, written once and never re-read -> don't pollute L2.
  float* op = out + (size_t)p * 32u;
#pragma unroll
  for (int i = 0; i < 8; ++i) {
    const f32x4 v = {resv[2 * i].x, resv[2 * i].y,
                     resv[2 * i + 1].x, resv[2 * i + 1].y};
    __builtin_nontemporal_store(v, (f32x4*)(op + 4 * i));
  }
}

extern "C" void kernel_launch(void* const* d_in, const int* in_sizes, int n_in,
                              void* d_out, int out_size, void* d_ws, size_t ws_size,
                              hipStream_t stream) {
  (void)in_sizes; (void)n_in; (void)out_size; (void)d_ws; (void)ws_size;
  const float* x      = (const float*)d_in[0];  // (NPTS, 3) float32
  const float* tables = (const float*)d_in[1];  // (16, 2^19, 2) float32
  float* out  = (float*)d_out;                  // (NPTS, 32) float32
  float* keep = out + (size_t)NPTS * 32u;       // (NPTS,) mask as 1.0/0.0

  hashgrid_fwd<<<NPTS / 256u, 256, 0, stream>>>(x, tables, out, keep);
}